// AdaptiveKernelModule_19542101196948
// MI455X (gfx1250) — compile-verified
//
#include <hip/hip_runtime.h>

typedef __attribute__((ext_vector_type(2))) float v2f;
typedef __attribute__((ext_vector_type(8))) float v8f;

#define N_     16
#define C_     128
#define CM_    32
#define H_     192
#define W_     192
#define HW_    (H_ * W_)
#define P_     128
#define TILE_H 16
#define TILE_W 30
#define FH_    18            // TILE_H + 2 halo rows
#define FW_    32            // 2 groups of 16 cols (TILE_W + 2 halo cols)
#define NPIX_  (TILE_H * TILE_W)

__device__ __forceinline__ int imin_(int a, int b) { return a < b ? a : b; }
__device__ __forceinline__ int imax_(int a, int b) { return a > b ? a : b; }

// ---------------------------------------------------------------------------
// Kernel 1: adaptive max pool x[N,C,192,192] -> xp[N,C,3,3] (64x64 windows)
// one block per (n,c); coalesced full-image scan, 9 running maxes per thread
// ---------------------------------------------------------------------------
__global__ __launch_bounds__(256) void pool_kernel(const float* __restrict__ x,
                                                   float* __restrict__ xp) {
  int nc = blockIdx.x;
  const float* xi = x + (size_t)nc * HW_;
  int tid = threadIdx.x;
  float mx[9];
#pragma unroll
  for (int i = 0; i < 9; ++i) mx[i] = -3.402823466e+38f;
  for (int idx = tid; idx < HW_; idx += 256) {
    int h = idx / W_;
    int w = idx - h * W_;
    int win = (h >> 6) * 3 + (w >> 6);
    float v = xi[idx];
    mx[win] = fmaxf(mx[win], v);
  }
  __shared__ float red[9][256];
#pragma unroll
  for (int i = 0; i < 9; ++i) red[i][tid] = mx[i];
  __syncthreads();
  for (int s = 128; s > 0; s >>= 1) {
    if (tid < s) {
#pragma unroll
      for (int i = 0; i < 9; ++i) red[i][tid] = fmaxf(red[i][tid], red[i][tid + s]);
    }
    __syncthreads();
  }
  if (tid < 9) xp[(size_t)nc * 9 + tid] = red[tid][0];
}

// ---------------------------------------------------------------------------
// Kernel 2: g[n,ci,kh,kw] = relu(sum_c W1[ci,c]*xp[n,c,kh,kw] + b1[ci])
// one block per n, 288 threads (32 ci x 9 positions)
// ---------------------------------------------------------------------------
__global__ __launch_bounds__(288) void g_kernel(const float* __restrict__ xp,
                                                const float* __restrict__ W1,
                                                const float* __restrict__ b1,
                                                float* __restrict__ g) {
  int n = blockIdx.x;
  int t = threadIdx.x;
  if (t < CM_ * 9) {
    int ci = t / 9;
    int kk = t - ci * 9;
    float s = b1[ci];
    const float* wr = W1 + ci * C_;
    const float* xr = xp + (size_t)n * C_ * 9 + kk;
    for (int c = 0; c < C_; ++c) s += wr[c] * xr[c * 9];
    g[(size_t)n * CM_ * 9 + t] = fmaxf(s, 0.0f);
  }
}

// ---------------------------------------------------------------------------
// Kernel 3: fold Wk/bk/b_adap/Wf/bf into per-output-channel affine (A,B,Cc)
// ---------------------------------------------------------------------------
__global__ __launch_bounds__(128) void abc_kernel(const float* __restrict__ Wk,
                                                  const float* __restrict__ bk,
                                                  const float* __restrict__ b_adap,
                                                  const float* __restrict__ Wf,
                                                  const float* __restrict__ bf,
                                                  float* __restrict__ Ax,
                                                  float* __restrict__ Bx,
                                                  float* __restrict__ Cx) {
  int p = threadIdx.x;
  float a = 0.f, b = 0.f, c = 0.f;
  const float* wr = Wf + p * CM_;
  for (int co = 0; co < CM_; ++co) {
    float w = wr[co];
    a += w * Wk[co];
    b += w * bk[co];
    c += w * b_adap[co];
  }
  Ax[p] = a;
  Bx[p] = b;
  Cx[p] = c + bf[p];
}

// ---------------------------------------------------------------------------
// Main fused kernel: conv1 (WMMA f32 16x16x4) -> S,T 3x3 reduce -> affine out
// grid (7 col-tiles, 12 row-tiles, 16 samples), block 256 (8 waves, wave32)
// ---------------------------------------------------------------------------
__global__ __launch_bounds__(256) void fused_kernel(const float* __restrict__ x,
                                                    const float* __restrict__ W1,
                                                    const float* __restrict__ b1,
                                                    const float* __restrict__ g,
                                                    const float* __restrict__ Ax,
                                                    const float* __restrict__ Bx,
                                                    const float* __restrict__ Cx,
                                                    float* __restrict__ out) {
  __shared__ float sW1[CM_ * C_];        // 16 KB: W1 staged once
  __shared__ float sF[16][FH_][FW_];     // 36 KB: f for one 16-channel half
  __shared__ float sG[CM_ * 9];          // per-sample adaptive weights
  __shared__ float sB1[CM_];
  __shared__ float sS[NPIX_];
  __shared__ float sT[NPIX_];
  __shared__ float sA[P_], sB[P_], sC[P_];

  const int n  = blockIdx.z;
  const int r0 = blockIdx.y * TILE_H;
  const int c0 = blockIdx.x * TILE_W;
  const int tid  = threadIdx.x;
  const int wave = tid >> 5;
  const int lane = tid & 31;
  const int m    = lane & 15;          // pixel / M row within 16-group
  const int hi   = (lane >= 16) ? 1 : 0;

  for (int i = tid; i < CM_ * C_; i += 256) sW1[i] = W1[i];
  for (int i = tid; i < CM_ * 9; i += 256) sG[i] = g[(size_t)n * CM_ * 9 + i];
  if (tid < CM_) sB1[tid] = b1[tid];
  if (tid < P_) { sA[tid] = Ax[tid]; sB[tid] = Bx[tid]; sC[tid] = Cx[tid]; }
  __syncthreads();

  const float* xn = x + (size_t)n * C_ * HW_;
  float Sacc[2] = {0.f, 0.f};
  float Tacc[2] = {0.f, 0.f};

  for (int half = 0; half < 2; ++half) {
    const int cobase = half * 16;
    // ---- f for this channel half over the 18x32 halo region (WMMA GEMM) ----
    for (int task = wave; task < 36; task += 8) {   // wave-uniform loop: EXEC full
      const int fr = task >> 1;                     // region row 0..17
      const int cg = task & 1;                      // col group 0/1
      const int gr = r0 + fr - 1;                   // global row (halo)
      const int gcol = c0 - 1 + cg * 16 + m;        // this lane's global col
      const bool pix_ok = (gr >= 0) & (gr < H_) & (gcol >= 0) & (gcol < W_);
      const int grc = imin_(imax_(gr, 0), H_ - 1);  // clamp (keep EXEC all-1)
      const int gcc = imin_(imax_(gcol, 0), W_ - 1);
      const float* xpix = xn + (size_t)grc * W_ + gcc;
      const int koff = hi ? 2 : 0;
      const float* wrow = sW1 + (cobase + m) * C_;

      v8f acc = {};
#pragma unroll 4
      for (int kc = 0; kc < 32; ++kc) {
        const int k = kc * 4 + koff;
        v2f a;                                      // A: W1[co, k..k+1]
        a.x = wrow[k];
        a.y = wrow[k + 1];
        v2f b;                                      // B: x[k..k+1, pixel]
        b.x = xpix[(size_t)k * HW_];
        b.y = xpix[(size_t)(k + 1) * HW_];
        acc = __builtin_amdgcn_wmma_f32_16x16x4_f32(
            false, a, false, b, (short)0, acc, false, false);
      }
      // D element (co_l = j + hi*8, pixel = m): bias + relu + zero-pad, to LDS
      const int colr = cg * 16 + m;
#pragma unroll
      for (int j = 0; j < 8; ++j) {
        const int co_l = j + (hi ? 8 : 0);
        float v = acc[j] + sB1[cobase + co_l];
        v = v > 0.f ? v : 0.f;
        if (!pix_ok) v = 0.f;                       // conv zero-padding
        sF[co_l][fr][colr] = v;
      }
    }
    __syncthreads();

    // ---- accumulate S (g-weighted 3x3x16) and T (plain 3x3x16) ----
#pragma unroll
    for (int pp = 0; pp < 2; ++pp) {
      const int pid = tid + pp * 256;
      if (pid < NPIX_) {
        const int pr = pid / TILE_W;
        const int pc = pid - pr * TILE_W;
        float s = 0.f, t = 0.f;
        for (int ci = 0; ci < 16; ++ci) {
          const float* grow = sG + (cobase + ci) * 9;
#pragma unroll
          for (int kh = 0; kh < 3; ++kh) {
#pragma unroll
            for (int kw = 0; kw < 3; ++kw) {
              const float fv = sF[ci][pr + kh][pc + kw];
              s = fmaf(grow[kh * 3 + kw], fv, s);
              t += fv;
            }
          }
        }
        Sacc[pp] += s;
        Tacc[pp] += t;
      }
    }
    __syncthreads();   // before sF is overwritten by the next half
  }

  // ---- publish S,T then stream the 128-channel affine output ----
#pragma unroll
  for (int pp = 0; pp < 2; ++pp) {
    const int pid = tid + pp * 256;
    if (pid < NPIX_) { sS[pid] = Sacc[pp]; sT[pid] = Tacc[pp]; }
  }
  __syncthreads();

  const int tw = imin_(TILE_W, W_ - c0);            // ragged last col tile
  float* outn = out + (size_t)n * P_ * HW_;
  for (int pi = 0; pi < 16; ++pi) {
    const int p = wave + pi * 8;
    const float a = sA[p], b = sB[p], cc = sC[p];
    float* op = outn + (size_t)p * HW_;
    for (int j = lane; j < NPIX_; j += 32) {
      const int pr = j / TILE_W;
      const int pc = j - pr * TILE_W;
      if (pc < tw) op[(size_t)(r0 + pr) * W_ + (c0 + pc)] = fmaf(a, sS[j], fmaf(b, sT[j], cc));
    }
  }
}

// ---------------------------------------------------------------------------
extern "C" void kernel_launch(void* const* d_in, const int* in_sizes, int n_in,
                              void* d_out, int out_size, void* d_ws, size_t ws_size,
                              hipStream_t stream) {
  const float* x      = (const float*)d_in[0];
  const float* W1     = (const float*)d_in[1];
  const float* b1     = (const float*)d_in[2];
  const float* Wk     = (const float*)d_in[3];
  const float* bk     = (const float*)d_in[4];
  const float* b_adap = (const float*)d_in[5];
  const float* Wf     = (const float*)d_in[6];
  const float* bf     = (const float*)d_in[7];
  float* out = (float*)d_out;

  float* ws = (float*)d_ws;
  float* xp = ws;                             // N*C*9     = 18432
  float* g  = ws + N_ * C_ * 9;               // N*CM*9    = 4608
  float* Ax = g + N_ * CM_ * 9;               // 128
  float* Bx = Ax + P_;                        // 128
  float* Cx = Bx + P_;                        // 128

  pool_kernel<<<N_ * C_, 256, 0, stream>>>(x, xp);
  g_kernel<<<N_, 288, 0, stream>>>(xp, W1, b1, g);
  abc_kernel<<<1, 128, 0, stream>>>(Wk, bk, b_adap, Wf, bf, Ax, Bx, Cx);

  dim3 grid((W_ + TILE_W - 1) / TILE_W, H_ / TILE_H, N_);  // 7 x 12 x 16
  fused_kernel<<<grid, 256, 0, stream>>>(x, W1, b1, g, Ax, Bx, Cx, out);
}